// GDER_2516850835459
// MI455X (gfx1250) — compile-verified
//
#include <hip/hip_runtime.h>
#include <hip/hip_bf16.h>

typedef float v2f  __attribute__((ext_vector_type(2)));
typedef float v8f  __attribute__((ext_vector_type(8)));
typedef unsigned int u32x4 __attribute__((ext_vector_type(4)));
typedef int   i32x8 __attribute__((ext_vector_type(8)));
typedef int   i32x4 __attribute__((ext_vector_type(4)));

#define IMG_DIM   512
#define OUT_DIM   498           // 512 - 15 + 1
#define TILE      16
#define IN_DIM    32            // 16 outputs + 14 halo, padded to 32
#define TILES_1D  32            // ceil(498/16)
#define TILES_IMG (TILES_1D * TILES_1D)   // 1024
#define N_IMG     64

// ws layout (floats): [0..1535] tabH, [1536..3071] tabV, [4096..] partials
#define TAB_H_OFF   0
#define TAB_V_OFF   1536
#define PART_OFF    4096

#if __has_builtin(__builtin_amdgcn_tensor_load_to_lds)
#define HAVE_TDM 1
#else
#define HAVE_TDM 0
#endif

// ---------------------------------------------------------------------------
// Init: build the WMMA Toeplitz fragment tables in double precision,
// replicating the reference's normalization quirks:
//   GxN = Gx / (sum(Gx) + 1e-7)   GyN = Gy / sum(Gy) + 1e-7
// Separable: Gx(y,x)=A*u(x)*v(y), Gy(y,x)=A*v(x)*u(y),
//   u(t)=-t*exp(-t^2/2s^2), v(t)=exp(-t^2/2s^2), A=1/(2*pi*s*s^2)
// tabH[f][s][lane] = stage-1 B fragment (f: 0=s_x*u, 1=v, 2=ones)
// tabV[f][s][lane] = stage-2 A fragment (f: 0=v, 1=s_y*u, 2=1e-7*ones)
// ---------------------------------------------------------------------------
__global__ void gder_init_tabs(float* ws) {
    const int tid = blockIdx.x * blockDim.x + threadIdx.x;  // 0..767
    if (tid >= 768) return;

    const double PI = 3.14159265358979323846;
    const double sig = 7.0 / 2.5;
    const double twoSig2 = 2.0 * sig * sig;
    const double Gc = 1.0 / (2.0 * PI * sig);
    double sGx = 0.0, sGy = 0.0;
    for (int yy = -7; yy <= 7; ++yy)        // row-major like numpy
        for (int xx = -7; xx <= 7; ++xx) {
            double G = exp(-(double)(xx * xx + yy * yy) / twoSig2) * Gc;
            sGx += -(double)xx * G / (sig * sig);
            sGy += -(double)yy * G / (sig * sig);
        }
    const double A   = Gc / (sig * sig);
    const double s_x = A / (sGx + 1e-7);    // Rx horizontal u-scale
    const double s_y = A / sGy;             // Ry vertical   u-scale (quirk)

    const int lane = tid & 31;
    const int s    = (tid >> 5) & 7;
    const int f    = tid >> 8;              // 0..2
    const int j    = lane & 15;
    const int p0   = 4 * s + 2 * (lane >> 4);

    #pragma unroll
    for (int e = 0; e < 2; ++e) {
        const int d = p0 + e - j;
        double h = 0.0, v = 0.0;
        if (d >= 0 && d <= 14) {
            const double t = (double)(d - 7);
            const double g = exp(-t * t / twoSig2);
            const double u = -t * g;
            h = (f == 0) ? s_x * u : (f == 1) ? g : 1.0;
            v = (f == 0) ? g : (f == 1) ? s_y * u : 1e-7;
        }
        ws[TAB_H_OFF + tid * 2 + e] = (float)h;
        ws[TAB_V_OFF + tid * 2 + e] = (float)v;
    }
}

// ---------------------------------------------------------------------------
// One wave = one 16x16 output tile. Separable conv as Toeplitz GEMMs on
// V_WMMA_F32_16X16X4_F32. Input halo staged into LDS by the Tensor Data
// Mover (OOB rows/cols zero-filled by TDM's tensor_dim clamp).
// ---------------------------------------------------------------------------
__global__ __launch_bounds__(32)
void gder_tile_kernel(const float* __restrict__ x,
                      const float* __restrict__ ws,
                      float* __restrict__ partials) {
    __shared__ float lds_in[IN_DIM * IN_DIM];      // 32x32 input halo tile
    __shared__ float lds_h[3][IN_DIM * TILE];      // 32x16 H panel per filter

    const int lane    = threadIdx.x;               // 0..31 (wave32)
    const int mrow    = lane & 15;
    const int halfsel = lane >> 4;
    const int tile_id = blockIdx.x;
    const int tx  = tile_id & (TILES_1D - 1);
    const int ty  = (tile_id >> 5) & (TILES_1D - 1);
    const int img = tile_id >> 10;
    const int x0 = tx * TILE;
    const int y0 = ty * TILE;

    const float* src  = x + (size_t)img * IMG_DIM * IMG_DIM;
    const float* tabH = ws + TAB_H_OFF;
    const float* tabV = ws + TAB_V_OFF;

    // ---- preload constant fragments (hot 12KB region, L2/WGP$ resident) ---
    v2f bH[3][8];
    v2f aV[3][8];
    #pragma unroll
    for (int f = 0; f < 3; ++f)
        #pragma unroll
        for (int s = 0; s < 8; ++s) {
            const int idx = ((f * 8 + s) * 32 + lane) * 2;
            bH[f][s] = *(const v2f*)(tabH + idx);
            aV[f][s] = *(const v2f*)(tabV + idx);
        }

#if HAVE_TDM
    // ---- TDM: DMA the 32x32 fp32 halo tile into LDS -----------------------
    // D# group0: count=1 | lds_addr | 57b global_addr | type=2
    // D# group1: data_size=4B, tensor_dim = remaining extent from tile start
    //            (rows/cols past the image edge read as zero), tile 32x32,
    //            row stride 512 elements.
    {
        const uint64_t gaddr = (uint64_t)(uintptr_t)(src + (size_t)y0 * IMG_DIM + x0);
        const uint32_t ldsa  = (uint32_t)(uintptr_t)(&lds_in[0]);
        const uint32_t td0   = (uint32_t)(IMG_DIM - x0);   // valid cols from start
        const uint32_t td1   = (uint32_t)(IMG_DIM - y0);   // valid rows from start

        u32x4 g0;
        g0.x = 1u;                                         // count=1, user desc
        g0.y = ldsa;                                       // lds_addr
        g0.z = (uint32_t)gaddr;                            // global_addr[31:0]
        g0.w = (uint32_t)(gaddr >> 32) | 0x80000000u;      // addr[56:32] | type=2

        i32x8 g1;
        g1[0] = 0x00020000;                                // data_size = 2 (4B)
        g1[1] = (int)((td0 & 0xFFFFu) << 16);              // tensor_dim0[15:0]
        g1[2] = (int)((td0 >> 16) | ((td1 & 0xFFFFu) << 16)); // td0 hi | td1 lo
        g1[3] = (int)((td1 >> 16) | (32u << 16));          // td1 hi | tile_dim0=32
        g1[4] = 32;                                        // tile_dim1=32, tile_dim2=0
        g1[5] = IMG_DIM;                                   // dim0_stride[31:0]
        g1[6] = 0;                                         // stride hi | d1stride lo
        g1[7] = 0;
        i32x4 gz4 = {0, 0, 0, 0};
        i32x8 gz8 = {0, 0, 0, 0, 0, 0, 0, 0};

        __builtin_amdgcn_tensor_load_to_lds(g0, g1, gz4, gz4, gz8, 0);
        __builtin_amdgcn_s_wait_tensorcnt(0);
    }
#else
    // ---- fallback: manual masked load of the halo tile --------------------
    {
        const int gy = y0 + lane;
        const bool rowok = gy < IMG_DIM;
        #pragma unroll
        for (int c = 0; c < IN_DIM; c += 4) {
            float4 val = make_float4(0.f, 0.f, 0.f, 0.f);
            const int gx = x0 + c;
            if (rowok) {
                if (gx + 3 < IMG_DIM) {
                    val = *(const float4*)(src + (size_t)gy * IMG_DIM + gx);
                } else {
                    float tmp[4] = {0.f, 0.f, 0.f, 0.f};
                    #pragma unroll
                    for (int e = 0; e < 4; ++e)
                        if (gx + e < IMG_DIM) tmp[e] = src[(size_t)gy * IMG_DIM + gx + e];
                    val = make_float4(tmp[0], tmp[1], tmp[2], tmp[3]);
                }
            }
            *(float4*)(&lds_in[lane * IN_DIM + c]) = val;
        }
    }
#endif
    __syncthreads();

    // ---- stage 1: horizontal Toeplitz GEMMs  H[r,j] = sum_p In[r,p]*h(p-j)
    #pragma unroll
    for (int rb = 0; rb < 2; ++rb) {
        v2f a[8];
        const int r = rb * 16 + mrow;              // A lanes: M = lane&15
        #pragma unroll
        for (int s = 0; s < 8; ++s) {
            const int p = 4 * s + 2 * halfsel;     // VGPR0: K=p, VGPR1: K=p+1
            a[s].x = lds_in[r * IN_DIM + p];
            a[s].y = lds_in[r * IN_DIM + p + 1];
        }
        #pragma unroll
        for (int f = 0; f < 3; ++f) {
            v8f acc = {};
            #pragma unroll
            for (int s = 0; s < 8; ++s)
                acc = __builtin_amdgcn_wmma_f32_16x16x4_f32(
                        false, a[s], false, bH[f][s], (short)0, acc, false, false);
            #pragma unroll
            for (int r2 = 0; r2 < 8; ++r2) {       // D: M = r2 + 8*halfsel
                const int row = rb * 16 + r2 + 8 * halfsel;
                lds_h[f][row * TILE + mrow] = acc[r2];
            }
        }
    }
    __syncthreads();

    // ---- stage 2: vertical Toeplitz GEMMs  Out[y,x] = sum_p w(p-y)*H[p,x]
    v8f rx = {};
    #pragma unroll
    for (int s = 0; s < 8; ++s) {
        const int p0 = 4 * s + 2 * halfsel;
        v2f b;                                     // B: H[p][n], n = mrow
        b.x = lds_h[0][p0 * TILE + mrow];
        b.y = lds_h[0][(p0 + 1) * TILE + mrow];
        rx = __builtin_amdgcn_wmma_f32_16x16x4_f32(
                false, aV[0][s], false, b, (short)0, rx, false, false);
    }
    v8f ry = {};
    #pragma unroll
    for (int f = 1; f < 3; ++f)                    // u-pass + box-pass accumulate
        #pragma unroll
        for (int s = 0; s < 8; ++s) {
            const int p0 = 4 * s + 2 * halfsel;
            v2f b;
            b.x = lds_h[f][p0 * TILE + mrow];
            b.y = lds_h[f][(p0 + 1) * TILE + mrow];
            ry = __builtin_amdgcn_wmma_f32_16x16x4_f32(
                    false, aV[f][s], false, b, (short)0, ry, false, false);
        }

    // ---- FM = Rx^2 + Ry^2, masked partial sum, wave32 reduction -----------
    float sum = 0.f;
    #pragma unroll
    for (int r2 = 0; r2 < 8; ++r2) {
        const int oy = y0 + r2 + 8 * halfsel;
        const int ox = x0 + mrow;
        const float fx = rx[r2], fy = ry[r2];
        const float fm = fx * fx + fy * fy;
        sum += (oy < OUT_DIM && ox < OUT_DIM) ? fm : 0.f;
    }
    #pragma unroll
    for (int off = 16; off > 0; off >>= 1)
        sum += __shfl_xor(sum, off, 32);
    if (lane == 0) partials[tile_id] = sum;
}

// ---------------------------------------------------------------------------
// Deterministic per-image reduction: 1024 tile partials -> mean
// ---------------------------------------------------------------------------
__global__ __launch_bounds__(256)
void gder_reduce(const float* __restrict__ partials, float* __restrict__ out) {
    __shared__ float red[256];
    const int img = blockIdx.x;
    const float* p = partials + img * TILES_IMG;
    float s = 0.f;
    for (int i = threadIdx.x; i < TILES_IMG; i += 256) s += p[i];
    red[threadIdx.x] = s;
    __syncthreads();
    for (int stride = 128; stride > 0; stride >>= 1) {
        if (threadIdx.x < stride) red[threadIdx.x] += red[threadIdx.x + stride];
        __syncthreads();
    }
    if (threadIdx.x == 0)
        out[img] = red[0] / ((float)OUT_DIM * (float)OUT_DIM);
}

extern "C" void kernel_launch(void* const* d_in, const int* in_sizes, int n_in,
                              void* d_out, int out_size, void* d_ws, size_t ws_size,
                              hipStream_t stream) {
    const float* x  = (const float*)d_in[0];
    float* ws       = (float*)d_ws;
    float* partials = (float*)d_ws + PART_OFF;
    float* out      = (float*)d_out;         // 64 floats ([64,1])

    gder_init_tabs<<<3, 256, 0, stream>>>(ws);
    gder_tile_kernel<<<N_IMG * TILES_IMG, 32, 0, stream>>>(x, ws, partials);
    gder_reduce<<<N_IMG, 256, 0, stream>>>(partials, out);
}